// SAGEClf_9560597201501
// MI455X (gfx1250) — compile-verified
//
#include <hip/hip_runtime.h>

#define D1 128   // input / hidden dim
#define D2 64    // second hidden dim
#define NC 10    // classes
#define BN_EPS 1e-5f

typedef float v2f __attribute__((ext_vector_type(2)));
typedef float v8f __attribute__((ext_vector_type(8)));

static __device__ __forceinline__ v8f wmma_f32(v2f a, v2f b, v8f c) {
    // V_WMMA_F32_16X16X4_F32 : D = A(16x4) * B(4x16) + C(16x16), all fp32
    return __builtin_amdgcn_wmma_f32_16x16x4_f32(
        /*neg_a=*/false, a, /*neg_b=*/false, b,
        /*c_mod=*/(short)0, c, /*reuse_a=*/false, /*reuse_b=*/false);
}

// ---------------------------------------------------------------- zero (16B chunks)
__global__ void k_zero(float* __restrict__ p, long n4) {
    long i = (long)blockIdx.x * blockDim.x + threadIdx.x;
    long stride = (long)gridDim.x * blockDim.x;
    float4 z = make_float4(0.f, 0.f, 0.f, 0.f);
    for (; i < n4; i += stride) ((float4*)p)[i] = z;
}

// ---------------------------------------------------------------- CSR build
__global__ void k_degree(const int* __restrict__ dst, int* __restrict__ cnt, int E) {
    int e = blockIdx.x * blockDim.x + threadIdx.x;
    if (e < E) atomicAdd(cnt + dst[e], 1);
}

// per-256-node block degree sums
__global__ void k_blocksum(const int* __restrict__ cnt, int* __restrict__ bsum, int N) {
    __shared__ int s[256];
    int i = blockIdx.x * 256 + threadIdx.x;
    s[threadIdx.x] = (i < N) ? cnt[i] : 0;
    __syncthreads();
    for (int off = 128; off > 0; off >>= 1) {
        if (threadIdx.x < off) s[threadIdx.x] += s[threadIdx.x + off];
        __syncthreads();
    }
    if (threadIdx.x == 0) bsum[blockIdx.x] = s[0];
}

// serial exclusive scan of block sums (NB ~ 391, trivial)
__global__ void k_scan_bsums(const int* __restrict__ bsum, int* __restrict__ boff,
                             int NB, int* __restrict__ row_ptr, int N) {
    if (blockIdx.x == 0 && threadIdx.x == 0) {
        int running = 0;
        for (int b = 0; b < NB; ++b) { boff[b] = running; running += bsum[b]; }
        row_ptr[N] = running;   // == E
    }
}

// per-block serial exclusive scan of 256 degrees -> row_ptr + cursor
__global__ void k_scan_local(const int* __restrict__ cnt, const int* __restrict__ boff,
                             int* __restrict__ row_ptr, int* __restrict__ cursor, int N) {
    if (threadIdx.x == 0) {
        int b = blockIdx.x;
        int running = boff[b];
        int base = b * 256;
        for (int j = 0; j < 256; ++j) {
            int i = base + j;
            if (i < N) {
                row_ptr[i] = running;
                cursor[i]  = running;
                running += cnt[i];
            }
        }
    }
}

// bucket src node ids by dst segment
__global__ void k_fill(const int* __restrict__ src, const int* __restrict__ dst,
                       int* __restrict__ cursor, int* __restrict__ col, int E) {
    int e = blockIdx.x * blockDim.x + threadIdx.x;
    if (e < E) {
        int pos = atomicAdd(cursor + dst[e], 1);
        col[pos] = src[e];
    }
}

// ---------------------------------------------------------------- pull-style mean agg
// wave per node, float4 per lane: agg[i,:] = mean_{s in N(i)} feat[s,:]  (dim 128)
__global__ __launch_bounds__(256) void k_gather_mean(const float* __restrict__ feat,
                                                     const int* __restrict__ row_ptr,
                                                     const int* __restrict__ col,
                                                     float* __restrict__ agg, int N) {
    int wid  = blockIdx.x * 8 + (threadIdx.x >> 5);
    int lane = threadIdx.x & 31;
    if (wid >= N) return;                       // uniform per wave
    int start = row_ptr[wid], end = row_ptr[wid + 1];
    float4 acc = make_float4(0.f, 0.f, 0.f, 0.f);
    const float* base = feat + lane * 4;
    for (int e = start; e < end; ++e) {
        int s = col[e];
        float4 v = *(const float4*)(base + (long)s * D1);
        acc.x += v.x; acc.y += v.y; acc.z += v.z; acc.w += v.w;
    }
    int d = end - start;
    float sc = 1.0f / (float)(d > 0 ? d : 1);
    acc.x *= sc; acc.y *= sc; acc.z *= sc; acc.w *= sc;
    *(float4*)(agg + (long)wid * D1 + lane * 4) = acc;
}

// ---------------------------------------------------------------- fused SAGE GEMM
// out = relu( BN( agg @ Wl + bl + xin @ Wr ) ),  BN in eval mode.
// Wave computes a 16 x NOUT tile via V_WMMA_F32_16X16X4_F32. Weight chunks are
// staged to LDS pair-swizzled so each lane's (k,k+1) B values are one ds_load_b64.
template <int K, int NOUT>
__global__ __launch_bounds__(256) void k_sage_gemm(
    const float* __restrict__ agg, const float* __restrict__ xin,
    const float* __restrict__ Wl,  const float* __restrict__ Wr,
    const float* __restrict__ bl,  const float* __restrict__ g,
    const float* __restrict__ be,  const float* __restrict__ mn,
    const float* __restrict__ vr,  float* __restrict__ out, int numTiles) {
    constexpr int NT   = NOUT / 16;   // output n-tiles per wave
    constexpr int CK   = 32;          // K rows staged per LDS chunk
    constexpr int LOGN = (NOUT == 128) ? 7 : 6;
    extern __shared__ float smem[];   // [CK*NOUT] Wl chunk | [CK*NOUT] Wr chunk (swizzled)
    float* sWl = smem;
    float* sWr = smem + CK * NOUT;

    const int tid  = threadIdx.x;
    const int wave = tid >> 5;
    const int lane = tid & 31;
    const int mrow = lane & 15;       // M (A) / N (B,C,D) within tile
    const int hi   = lane >> 4;       // selects K pair (A/B) / row half (C/D)

    int tile = blockIdx.x * 8 + wave;
    if (tile >= numTiles) tile = numTiles - 1;  // duplicate same-value work; EXEC all-1s
    const int row0 = tile * 16;
    const long rowOff = (long)(row0 + mrow) * K;

    v8f acc[NT];
    v8f zero = {0.f, 0.f, 0.f, 0.f, 0.f, 0.f, 0.f, 0.f};
#pragma unroll
    for (int t = 0; t < NT; ++t) acc[t] = zero;

    for (int kc = 0; kc < K; kc += CK) {
        __syncthreads();
        // pair-swizzled staging: sW[((k>>1)*NOUT + n)*2 + (k&1)] = W[kc+k][n]
        for (int i = tid; i < CK * NOUT; i += 256) {
            int k = i >> LOGN;
            int n = i & (NOUT - 1);
            int d = ((((k >> 1) << LOGN) + n) << 1) | (k & 1);
            sWl[d] = Wl[kc * NOUT + i];
            sWr[d] = Wr[kc * NOUT + i];
        }
        __syncthreads();
#pragma unroll
        for (int ks = 0; ks < CK / 4; ++ks) {
            const int kg = kc + ks * 4 + hi * 2;   // this lane's global K pair
            // A fragments: A[m=lane%16][k = (lane/16)*2 + v]
            v2f a1 = *(const v2f*)(agg + rowOff + kg);
            v2f a2 = *(const v2f*)(xin + rowOff + kg);
            const int pbase = ((ks * 2 + hi) << LOGN) << 1;  // pair-row base in LDS
#pragma unroll
            for (int nt = 0; nt < NT; ++nt) {
                const int n2 = (nt * 16 + mrow) << 1;
                v2f b1 = *(const v2f*)(sWl + pbase + n2);  // {W[k][n], W[k+1][n]}
                v2f b2 = *(const v2f*)(sWr + pbase + n2);
                acc[nt] = wmma_f32(a1, b1, acc[nt]);       // agg @ Wl
                acc[nt] = wmma_f32(a2, b2, acc[nt]);       // + x @ Wr
            }
        }
    }

    // Epilogue: +bias, eval BatchNorm, ReLU. D layout: lane owns col n=lane%16,
    // rows r + 8*(lane/16) for VGPR r = 0..7.
#pragma unroll
    for (int nt = 0; nt < NT; ++nt) {
        const int c = nt * 16 + mrow;
        const float alpha = g[c] * rsqrtf(vr[c] + BN_EPS);
        const float beta  = (bl[c] - mn[c]) * alpha + be[c];
#pragma unroll
        for (int r = 0; r < 8; ++r) {
            float val = fmaf(acc[nt][r], alpha, beta);
            val = fmaxf(val, 0.0f);
            out[(long)(row0 + r + 8 * hi) * NOUT + c] = val;
        }
    }
}

// ---------------------------------------------------------------- classifier head
__global__ void k_head(const float* __restrict__ h, const float* __restrict__ Wh,
                       const float* __restrict__ bh, float* __restrict__ out, int n) {
    __shared__ float sW[D2 * NC + NC];
    for (int i = threadIdx.x; i < D2 * NC + NC; i += blockDim.x)
        sW[i] = (i < D2 * NC) ? Wh[i] : bh[i - D2 * NC];
    __syncthreads();
    int node = blockIdx.x * blockDim.x + threadIdx.x;
    if (node >= n) return;
    const float* row = h + (long)node * D2;
    float acc[NC];
#pragma unroll
    for (int c = 0; c < NC; ++c) acc[c] = sW[D2 * NC + c];
    for (int k = 0; k < D2; ++k) {
        float xv = row[k];
#pragma unroll
        for (int c = 0; c < NC; ++c) acc[c] = fmaf(xv, sW[k * NC + c], acc[c]);
    }
#pragma unroll
    for (int c = 0; c < NC; ++c) out[(long)node * NC + c] = acc[c];
}

// ----------------------------------------------------------------
extern "C" void kernel_launch(void* const* d_in, const int* in_sizes, int n_in,
                              void* d_out, int out_size, void* d_ws, size_t ws_size,
                              hipStream_t stream) {
    const float* x   = (const float*)d_in[0];
    const int*   ei  = (const int*)d_in[1];
    const float* W1l = (const float*)d_in[2];
    const float* b1l = (const float*)d_in[3];
    const float* W1r = (const float*)d_in[4];
    const float* g1  = (const float*)d_in[5];
    const float* be1 = (const float*)d_in[6];
    const float* m1  = (const float*)d_in[7];
    const float* v1  = (const float*)d_in[8];
    const float* W2l = (const float*)d_in[9];
    const float* b2l = (const float*)d_in[10];
    const float* W2r = (const float*)d_in[11];
    const float* g2  = (const float*)d_in[12];
    const float* be2 = (const float*)d_in[13];
    const float* m2  = (const float*)d_in[14];
    const float* v2  = (const float*)d_in[15];
    const float* Wh  = (const float*)d_in[16];
    const float* bh  = (const float*)d_in[17];
    float* outp = (float*)d_out;

    const int N = in_sizes[0] / D1;   // 100000
    const int E = in_sizes[1] / 2;    // 1600000
    const int NB = (N + 255) / 256;   // 391 scan blocks
    const int* src = ei;
    const int* dst = ei + E;

    // Workspace: floats [ agg(N*128) | h1(N*128) | h2(N*64) ] then ints
    // [ cnt(N) | row_ptr(N+1) | cursor(N) | bsum(NB) | boff(NB) | col(E) ]
    float* ws = (float*)d_ws;
    const long FN = (long)N * D1;
    float* agg = ws;
    float* h1  = ws + FN;
    float* h2  = ws + 2 * FN;
    int* ip      = (int*)(ws + 2 * FN + (long)N * D2);
    int* cnt     = ip;
    int* row_ptr = cnt + N;
    int* cursor  = row_ptr + N + 1;
    int* bsum    = cursor + N;
    int* boff    = bsum + NB;
    int* col     = boff + NB;

    // ---- CSR build (once per call; reused by both layers) ----
    k_zero<<<256, 256, 0, stream>>>((float*)cnt, N / 4);
    k_degree<<<(E + 255) / 256, 256, 0, stream>>>(dst, cnt, E);
    k_blocksum<<<NB, 256, 0, stream>>>(cnt, bsum, N);
    k_scan_bsums<<<1, 32, 0, stream>>>(bsum, boff, NB, row_ptr, N);
    k_scan_local<<<NB, 32, 0, stream>>>(cnt, boff, row_ptr, cursor, N);
    k_fill<<<(E + 255) / 256, 256, 0, stream>>>(src, dst, cursor, col, E);

    const int aggBlocks  = (N + 7) / 8;       // wave per node
    const int numTiles   = N / 16;            // 6250
    const int gemmBlocks = (numTiles + 7) / 8;

    // ---- layer 1 ----
    k_gather_mean<<<aggBlocks, 256, 0, stream>>>(x, row_ptr, col, agg, N);
    {
        size_t shm = 2u * 32 * D1 * sizeof(float);  // 32 KB
        k_sage_gemm<D1, D1><<<gemmBlocks, 256, shm, stream>>>(
            agg, x, W1l, W1r, b1l, g1, be1, m1, v1, h1, numTiles);
    }

    // ---- layer 2 (agg buffer reused) ----
    k_gather_mean<<<aggBlocks, 256, 0, stream>>>(h1, row_ptr, col, agg, N);
    {
        size_t shm = 2u * 32 * D2 * sizeof(float);  // 16 KB
        k_sage_gemm<D1, D2><<<gemmBlocks, 256, shm, stream>>>(
            agg, h1, W2l, W2r, b2l, g2, be2, m2, v2, h2, numTiles);
    }

    // ---- head ----
    k_head<<<(N + 255) / 256, 256, 0, stream>>>(h2, Wh, bh, outp, N);
}